// _NonLocal0_75909251989680
// MI455X (gfx1250) — compile-verified
//
#include <hip/hip_runtime.h>

typedef float v2f __attribute__((ext_vector_type(2)));
typedef float v8f __attribute__((ext_vector_type(8)));

#define B_    8
#define C_    64
#define W_    64
#define H_    64
#define Z_    32
#define IC_   32
#define CS_   131072u     // W*H*Z (channel stride in x/out)
#define BS_   8388608u    // C*CS  (batch stride)
#define NPIX_ 32768       // B*W*H
#define NBLK1 1024
#define PPW_  8           // pixels per wave (1024 blocks * 4 waves * 8 = 32768)

// ---- LDS layout (float offsets) ----
#define L_WCATT 0               // WcatT[64][96]  (Wcat = [Wt;Wp;Wg] rows 0..95)
#define L_WWT   6144            // WwT[32][64]
#define L_BIAS  8192            // biasCat[96] = [bt,bp,bg]
#define L_BW    8288            // bw[64]
#define L_STAT  8352            // per-wave stats: 4 x (sum[64], ssq[64]) = 512
#define L_WAVE0 8864
#define L_WSTRIDE 5632
#define L_P   0                 // P[96][32]
#define L_PHI 3072              // phi_p[32][16]
#define L_G   3584              // g_p[32][16]
#define L_S   4096              // S[32][16]  (F then softmax)
#define L_YT  4608              // Yt[32][32]
#define LDS_FLOATS (L_WAVE0 + 4*L_WSTRIDE)   // 31392 floats = 125568 bytes

__device__ __forceinline__ v8f wmma4(v2f a, v2f b, v8f c) {
  // D = A(16x4,f32) * B(4x16,f32) + C(16x16,f32)
  return __builtin_amdgcn_wmma_f32_16x16x4_f32(false, a, false, b, (short)0, c,
                                               false, false);
}

// Intra-wave LDS fence: DS ops from one wave execute in order in the LDS pipe,
// so draining DScnt makes all prior lane-writes visible to any lane's reads.
// The "memory" clobber pins compiler-level ordering of the LDS accesses.
__device__ __forceinline__ void wave_lds_fence() {
  asm volatile("s_wait_dscnt 0x0" ::: "memory");
}

__global__ void nl_fused(const float* __restrict__ x,
                         const float* __restrict__ Wt, const float* __restrict__ bt,
                         const float* __restrict__ Wp, const float* __restrict__ bp,
                         const float* __restrict__ Wg, const float* __restrict__ bg,
                         const float* __restrict__ Ww, const float* __restrict__ bw,
                         float* __restrict__ out,
                         float* __restrict__ partial) {
  extern __shared__ float lds[];
  const int tid  = threadIdx.x;
  const int lane = tid & 31;
  const int wave = tid >> 5;
  const int half = lane >> 4;     // 0: lanes 0-15, 1: lanes 16-31
  const int l16  = lane & 15;

  // ---- stage weights (transposed) + biases, zero per-wave stats ----
  for (int i = tid; i < 96 * 64; i += 128) {
    int m = i >> 6, c = i & 63;
    float v = (m < 32) ? Wt[m * 64 + c]
            : (m < 64) ? Wp[(m - 32) * 64 + c]
                       : Wg[(m - 64) * 64 + c];
    lds[L_WCATT + c * 96 + m] = v;
  }
  for (int i = tid; i < 64 * 32; i += 128) {
    int cc = i >> 5, ic = i & 31;
    lds[L_WWT + ic * 64 + cc] = Ww[cc * 32 + ic];
  }
  if (tid < 96) lds[L_BIAS + tid] = (tid < 32) ? bt[tid] : (tid < 64) ? bp[tid - 32] : bg[tid - 64];
  if (tid < 64) lds[L_BW + tid] = bw[tid];
  for (int i = tid; i < 512; i += 128) lds[L_STAT + i] = 0.0f;
  __syncthreads();

  float* wl    = lds + L_WAVE0 + wave * L_WSTRIDE;
  float* wstat = lds + L_STAT + wave * 128;
  const int g = blockIdx.x * 4 + wave;

  for (int pi = 0; pi < PPW_; ++pi) {
    const int p   = g * PPW_ + pi;
    const int b   = p >> 12;
    const int rem = p & 4095;
    const int w   = rem >> 6;
    const int h   = rem & 63;
    const size_t base = (size_t)b * BS_ + (size_t)w * (H_ * Z_) + (size_t)h * Z_;
    const float* xb = x + base;

    if (p + 1 < NPIX_) {   // warm next pixel's x (global_prefetch_b8)
      int pn = p + 1, bn = pn >> 12, rn = pn & 4095;
      const float* nb = x + (size_t)bn * BS_ + (size_t)(rn >> 6) * (H_ * Z_) + (size_t)(rn & 63) * Z_;
      __builtin_prefetch((const void*)nb, 0, 3);
    }

    // ---- P = Wcat(96x64) @ X(64x32) + bias ;  x read exactly once, coalesced ----
    for (int nt = 0; nt < 2; ++nt) {
      float bf0[16], bf1[16];
      const int zcol = nt * 16 + l16;
#pragma unroll
      for (int k = 0; k < 16; ++k) {
        const int c0 = 4 * k + 2 * half;
        bf0[k] = xb[(size_t)c0 * CS_ + zcol];
        bf1[k] = xb[(size_t)(c0 + 1) * CS_ + zcol];
      }
      for (int mt = 0; mt < 6; ++mt) {
        // preload all A-fragments, then drain the WMMA chain back-to-back
        float a0[16], a1[16];
#pragma unroll
        for (int k = 0; k < 16; ++k) {
          const int kc = 4 * k + 2 * half;
          a0[k] = lds[L_WCATT + kc * 96 + mt * 16 + l16];
          a1[k] = lds[L_WCATT + (kc + 1) * 96 + mt * 16 + l16];
        }
        v8f acc = {0.f, 0.f, 0.f, 0.f, 0.f, 0.f, 0.f, 0.f};
#pragma unroll
        for (int k = 0; k < 16; ++k) {
          v2f a = {a0[k], a1[k]};
          v2f bb = {bf0[k], bf1[k]};
          acc = wmma4(a, bb, acc);
        }
#pragma unroll
        for (int r = 0; r < 8; ++r) {
          const int row = mt * 16 + half * 8 + r;
          wl[L_P + row * 32 + nt * 16 + l16] = acc[r] + lds[L_BIAS + row];
        }
      }
    }
    wave_lds_fence();

    // ---- max-pool z2 for phi (rows 32..63) and g (rows 64..95) ----
    for (int t = lane; t < 512; t += 32) {
      const int ic = t >> 4, nz = t & 15;
      wl[L_PHI + ic * 16 + nz] = fmaxf(wl[L_P + (32 + ic) * 32 + 2 * nz],
                                       wl[L_P + (32 + ic) * 32 + 2 * nz + 1]);
      wl[L_G + ic * 16 + nz]   = fmaxf(wl[L_P + (64 + ic) * 32 + 2 * nz],
                                       wl[L_P + (64 + ic) * 32 + 2 * nz + 1]);
    }
    wave_lds_fence();

    // ---- F = theta^T(32x32) @ phi(32x16) ----
    {
      float b0v[8], b1v[8];
#pragma unroll
      for (int k = 0; k < 8; ++k) {   // B (phi) shared across both m-tiles
        const int kc = 4 * k + 2 * half;
        b0v[k] = wl[L_PHI + kc * 16 + l16];
        b1v[k] = wl[L_PHI + (kc + 1) * 16 + l16];
      }
      for (int mt = 0; mt < 2; ++mt) {
        float a0[8], a1[8];
#pragma unroll
        for (int k = 0; k < 8; ++k) {
          const int kc = 4 * k + 2 * half;
          a0[k] = wl[L_P + kc * 32 + mt * 16 + l16];
          a1[k] = wl[L_P + (kc + 1) * 32 + mt * 16 + l16];
        }
        v8f acc = {0.f, 0.f, 0.f, 0.f, 0.f, 0.f, 0.f, 0.f};
#pragma unroll
        for (int k = 0; k < 8; ++k) {
          v2f a = {a0[k], a1[k]};
          v2f bb = {b0v[k], b1v[k]};
          acc = wmma4(a, bb, acc);
        }
#pragma unroll
        for (int r = 0; r < 8; ++r)
          wl[L_S + (mt * 16 + half * 8 + r) * 16 + l16] = acc[r];
      }
    }
    wave_lds_fence();

    // ---- softmax over nz: one row per lane ----
    {
      const int row = lane;
      float v[16], mx = -1e30f;
#pragma unroll
      for (int j = 0; j < 16; ++j) { v[j] = wl[L_S + row * 16 + j]; mx = fmaxf(mx, v[j]); }
      float s = 0.f;
#pragma unroll
      for (int j = 0; j < 16; ++j) { v[j] = __expf(v[j] - mx); s += v[j]; }
      const float inv = 1.0f / s;
#pragma unroll
      for (int j = 0; j < 16; ++j) wl[L_S + row * 16 + j] = v[j] * inv;
    }
    wave_lds_fence();

    // ---- Yt = g_p(32x16) @ S^T(16x32) ----
    for (int mt = 0; mt < 2; ++mt) {
      float a0[4], a1[4];
#pragma unroll
      for (int k = 0; k < 4; ++k) {   // A (g_p) shared across both n-tiles
        const int kc = 4 * k + 2 * half;
        a0[k] = wl[L_G + (mt * 16 + l16) * 16 + kc];
        a1[k] = wl[L_G + (mt * 16 + l16) * 16 + kc + 1];
      }
      for (int nt = 0; nt < 2; ++nt) {
        float b0v[4], b1v[4];
#pragma unroll
        for (int k = 0; k < 4; ++k) {
          const int kc = 4 * k + 2 * half;
          b0v[k] = wl[L_S + (nt * 16 + l16) * 16 + kc];
          b1v[k] = wl[L_S + (nt * 16 + l16) * 16 + kc + 1];
        }
        v8f acc = {0.f, 0.f, 0.f, 0.f, 0.f, 0.f, 0.f, 0.f};
#pragma unroll
        for (int k = 0; k < 4; ++k) {
          v2f a = {a0[k], a1[k]};
          v2f bb = {b0v[k], b1v[k]};
          acc = wmma4(a, bb, acc);
        }
#pragma unroll
        for (int r = 0; r < 8; ++r)
          wl[L_YT + (mt * 16 + half * 8 + r) * 32 + nt * 16 + l16] = acc[r];
      }
    }
    wave_lds_fence();

    // ---- WY = Ww(64x32) @ Yt(32x32) + bw ; store + per-wave channel stats ----
    for (int mt = 0; mt < 4; ++mt) {
      float a0[8], a1[8];
#pragma unroll
      for (int k = 0; k < 8; ++k) {   // A (Ww) shared across both n-tiles
        const int kc = 4 * k + 2 * half;
        a0[k] = lds[L_WWT + kc * 64 + mt * 16 + l16];
        a1[k] = lds[L_WWT + (kc + 1) * 64 + mt * 16 + l16];
      }
      for (int nt = 0; nt < 2; ++nt) {
        float b0v[8], b1v[8];
#pragma unroll
        for (int k = 0; k < 8; ++k) {
          const int kc = 4 * k + 2 * half;
          b0v[k] = wl[L_YT + kc * 32 + nt * 16 + l16];
          b1v[k] = wl[L_YT + (kc + 1) * 32 + nt * 16 + l16];
        }
        v8f acc = {0.f, 0.f, 0.f, 0.f, 0.f, 0.f, 0.f, 0.f};
#pragma unroll
        for (int k = 0; k < 8; ++k) {
          v2f a = {a0[k], a1[k]};
          v2f bb = {b0v[k], b1v[k]};
          acc = wmma4(a, bb, acc);
        }
#pragma unroll
        for (int r = 0; r < 8; ++r) {
          const int cch = mt * 16 + half * 8 + r;
          const float val = acc[r] + lds[L_BW + cch];
          out[base + (size_t)cch * CS_ + nt * 16 + l16] = val;
          float s1 = val, s2 = val * val;
#pragma unroll
          for (int off = 1; off < 16; off <<= 1) {
            s1 += __shfl_xor(s1, off, 32);
            s2 += __shfl_xor(s2, off, 32);
          }
          if (l16 == 0) {   // lanes 0 and 16 hit distinct channels: race-free
            wstat[cch]      += s1;
            wstat[64 + cch] += s2;
          }
        }
      }
    }
    wave_lds_fence();
  }

  // ---- deterministic per-block partials (sum the 4 per-wave regions) ----
  __syncthreads();
  partial[blockIdx.x * 128 + tid] =
      lds[L_STAT + tid] + lds[L_STAT + 128 + tid] +
      lds[L_STAT + 256 + tid] + lds[L_STAT + 384 + tid];
}

__global__ void nl_stats(const float* __restrict__ partial,
                         const float* __restrict__ gamma,
                         const float* __restrict__ beta,
                         float* __restrict__ stats) {
  __shared__ float red[128];
  const int t = threadIdx.x;
  float s = 0.0f;
  for (int b = 0; b < NBLK1; ++b) s += partial[b * 128 + t];
  red[t] = s;
  __syncthreads();
  if (t < 64) {
    const float Ninv = 1.0f / 1048576.0f;   // B*W*H*Z per channel
    const float mean = red[t] * Ninv;
    const float var  = red[64 + t] * Ninv - mean * mean;  // biased var (ddof=0)
    const float inv  = rsqrtf(var + 1e-5f);
    const float sc   = gamma[t] * inv;
    stats[t]      = sc;
    stats[64 + t] = beta[t] - mean * sc;
  }
}

__global__ void nl_bn_res(float* __restrict__ out,
                          const float* __restrict__ x,
                          const float* __restrict__ stats) {
  const size_t i4 = (size_t)blockIdx.x * blockDim.x + threadIdx.x;
  const size_t idx = i4 * 4;
  const int cch = (int)((idx >> 17) & 63);  // CS_ = 2^17, C_ = 64
  const float sc = stats[cch];
  const float sh = stats[64 + cch];
  float4 wy = ((const float4*)out)[i4];
  float4 xv = ((const float4*)x)[i4];
  float4 r;
  r.x = fmaf(wy.x, sc, sh) + xv.x;
  r.y = fmaf(wy.y, sc, sh) + xv.y;
  r.z = fmaf(wy.z, sc, sh) + xv.z;
  r.w = fmaf(wy.w, sc, sh) + xv.w;
  ((float4*)out)[i4] = r;
}

extern "C" void kernel_launch(void* const* d_in, const int* in_sizes, int n_in,
                              void* d_out, int out_size, void* d_ws, size_t ws_size,
                              hipStream_t stream) {
  (void)in_sizes; (void)n_in; (void)out_size; (void)ws_size;
  const float* x     = (const float*)d_in[0];
  // d_in[1] = spacings (unused by reference)
  const float* Wg    = (const float*)d_in[2];
  const float* bg    = (const float*)d_in[3];
  const float* Wt    = (const float*)d_in[4];
  const float* bt    = (const float*)d_in[5];
  const float* Wp    = (const float*)d_in[6];
  const float* bp    = (const float*)d_in[7];
  const float* Ww    = (const float*)d_in[8];
  const float* bw    = (const float*)d_in[9];
  const float* gamma = (const float*)d_in[10];
  const float* beta  = (const float*)d_in[11];

  float* out     = (float*)d_out;
  float* partial = (float*)d_ws;               // NBLK1*128 floats (512 KB)
  float* stats   = partial + NBLK1 * 128;      // 128 floats

  const size_t smem = (size_t)LDS_FLOATS * sizeof(float);  // 125568 B
  nl_fused<<<dim3(NBLK1), dim3(128), smem, stream>>>(
      x, Wt, bt, Wp, bp, Wg, bg, Ww, bw, out, partial);
  nl_stats<<<dim3(1), dim3(128), 0, stream>>>(partial, gamma, beta, stats);
  nl_bn_res<<<dim3(65536), dim3(256), 0, stream>>>(out, x, stats);
}